// get_model_60172491817899
// MI455X (gfx1250) — compile-verified
//
#include <hip/hip_runtime.h>
#include <stdint.h>

typedef __attribute__((ext_vector_type(2))) float v2f;
typedef __attribute__((ext_vector_type(8))) float v8f;

#define BB 32
#define NN 4096
#define MM 4096
#define ROWS_PER_BLOCK 128   // 8 waves * 16 rows
#define BLOCK_THREADS 256
#define NWAVES 8
#define TILE_UNROLL 4        // 4 x 16 = 64 columns per sync round

// Order-preserving float -> uint key (exact for all floats, incl. tiny negatives
// from the ||a||^2+||b||^2-2ab expansion). 0xFFFFFFFF is the +max sentinel.
__device__ __forceinline__ uint32_t f2key(float f) {
  uint32_t u = __float_as_uint(f);
  return (u & 0x80000000u) ? ~u : (u | 0x80000000u);
}
__device__ __forceinline__ float key2f(uint32_t k) {
  uint32_t u = (k & 0x80000000u) ? (k & 0x7FFFFFFFu) : ~k;
  return __uint_as_float(u);
}

__global__ void chamfer_init_colmin(uint32_t* __restrict__ colmin) {
  int i = blockIdx.x * blockDim.x + threadIdx.x;
  if (i < BB * MM) colmin[i] = 0xFFFFFFFFu;
}

__global__ __launch_bounds__(BLOCK_THREADS)
void chamfer_tiles(const float* __restrict__ x, const float* __restrict__ recon,
                   float* __restrict__ rowmin, uint32_t* __restrict__ colmin) {
  // recon staged once per block: {r0, r1, r2, ||r||^2} per row -> ds_load_b128
  __shared__ float4 lds_recon4[MM];                 // 64 KB
  __shared__ float  lds_colred[NWAVES * 64];        // 2 KB

  const int tid  = threadIdx.x;
  const int wave = tid >> 5;
  const int lane = tid & 31;
  const int half = lane >> 4;   // 0: K=0,1 / rows v ; 1: K=2,3 / rows v+8
  const int lr   = lane & 15;

  const int b      = blockIdx.x / (NN / ROWS_PER_BLOCK);
  const int rowBlk = blockIdx.x % (NN / ROWS_PER_BLOCK);
  const int nrow   = rowBlk * ROWS_PER_BLOCK + wave * 16;

  const float* xb = x     + (size_t)b * NN * 3;
  const float* rb = recon + (size_t)b * MM * 3;

  // ---- Stage recon batch (48 KB) + norms into LDS, cooperative ----
  for (int m = tid; m < MM; m += BLOCK_THREADS) {
    const float* rrow = rb + (size_t)m * 3;
    float r0 = rrow[0], r1 = rrow[1], r2 = rrow[2];
    float rr = fmaf(r2, r2, fmaf(r1, r1, r0 * r0));
    lds_recon4[m] = make_float4(r0, r1, r2, rr);
  }

  // ---- A operand (16x4 f32, K=3 padded to 4 with zeros), loop invariant ----
  v2f A;
  {
    const float* xr = xb + (size_t)(nrow + lr) * 3;
    float x0 = xr[0], x1 = xr[1], x2 = xr[2];
    A.x = half ? x2 : x0;
    A.y = half ? 0.0f : x1;
  }

  // Per-lane squared norms of the 8 x-rows this lane's D registers map to.
  float xx[8];
#pragma unroll
  for (int v = 0; v < 8; ++v) {
    const float* xr = xb + (size_t)(nrow + half * 8 + v) * 3;
    float x0 = xr[0], x1 = xr[1], x2 = xr[2];
    xx[v] = fmaf(x2, x2, fmaf(x1, x1, x0 * x0));
  }

  float rmin[8];
#pragma unroll
  for (int v = 0; v < 8; ++v) rmin[v] = 3.4e38f;

  __syncthreads();  // recon staging complete

  // ---- Sweep all M columns: 4 WMMA tiles per sync round ----
  for (int grp = 0; grp < MM / (16 * TILE_UNROLL); ++grp) {
    float cmin[TILE_UNROLL];

#pragma unroll
    for (int t = 0; t < TILE_UNROLL; ++t) {
      const int m0 = grp * (16 * TILE_UNROLL) + t * 16;
      float4 rv = lds_recon4[m0 + lr];   // ds_load_b128, conflict-free

      v2f Bv;
      Bv.x = half ? rv.z : rv.x;
      Bv.y = half ? 0.0f : rv.y;

      v8f c = {};
      c = __builtin_amdgcn_wmma_f32_16x16x4_f32(
          /*neg_a=*/false, A, /*neg_b=*/false, Bv,
          /*c_mod=*/(short)0, c, /*reuse_a=*/false, /*reuse_b=*/false);

      float cm = 3.4e38f;
#pragma unroll
      for (int v = 0; v < 8; ++v) {
        float d = fmaf(-2.0f, c[v], xx[v] + rv.w);
        rmin[v] = fminf(rmin[v], d);
        cm      = fminf(cm, d);
      }
      // combine the two 16-lane halves (rows v and v+8 of this strip)
      cm = fminf(cm, __shfl_xor(cm, 16, 32));
      cmin[t] = cm;
    }

    // ---- Cross-wave column-min: plain LDS stores + serial min over 8 waves ----
    if (half == 0) {
#pragma unroll
      for (int t = 0; t < TILE_UNROLL; ++t)
        lds_colred[wave * 64 + t * 16 + lr] = cmin[t];
    }
    __syncthreads();
    if (tid < 64) {
      float m = lds_colred[tid];
#pragma unroll
      for (int w = 1; w < NWAVES; ++w)
        m = fminf(m, lds_colred[w * 64 + tid]);
      atomicMin(&colmin[(size_t)b * MM + grp * 64 + tid], f2key(m));
    }
    __syncthreads();
  }

  // ---- Row-min reduce across the 16 lanes of each half ----
#pragma unroll
  for (int v = 0; v < 8; ++v) {
    float r = rmin[v];
    r = fminf(r, __shfl_xor(r, 1, 32));
    r = fminf(r, __shfl_xor(r, 2, 32));
    r = fminf(r, __shfl_xor(r, 4, 32));
    r = fminf(r, __shfl_xor(r, 8, 32));
    rmin[v] = r;
  }
  if (lr == 0) {
#pragma unroll
    for (int v = 0; v < 8; ++v)
      rowmin[(size_t)b * NN + nrow + half * 8 + v] = rmin[v];
  }
}

__global__ __launch_bounds__(256)
void chamfer_reduce(const float* __restrict__ rowmin,
                    const uint32_t* __restrict__ colmin,
                    float* __restrict__ out) {
  const int tid = threadIdx.x;
  double s_row = 0.0, s_col = 0.0;
  for (int i = tid; i < BB * NN; i += 256) s_row += (double)rowmin[i];
  for (int i = tid; i < BB * MM; i += 256) s_col += (double)key2f(colmin[i]);
  double s = s_row / (double)(BB * NN) + s_col / (double)(BB * MM);

  __shared__ double red[256];
  red[tid] = s;
  __syncthreads();
  for (int off = 128; off > 0; off >>= 1) {
    if (tid < off) red[tid] += red[tid + off];
    __syncthreads();
  }
  if (tid == 0) out[0] = (float)red[0];
}

extern "C" void kernel_launch(void* const* d_in, const int* in_sizes, int n_in,
                              void* d_out, int out_size, void* d_ws, size_t ws_size,
                              hipStream_t stream) {
  const float* x     = (const float*)d_in[0];   // [B,N,3] f32
  const float* recon = (const float*)d_in[1];   // [B,M,3] f32
  float* out = (float*)d_out;                   // scalar loss

  float*    rowmin = (float*)d_ws;                                   // B*N f32
  uint32_t* colmin = (uint32_t*)((char*)d_ws + (size_t)BB * NN * 4); // B*M u32

  chamfer_init_colmin<<<(BB * MM + 255) / 256, 256, 0, stream>>>(colmin);
  chamfer_tiles<<<BB * (NN / ROWS_PER_BLOCK), BLOCK_THREADS, 0, stream>>>(
      x, recon, rowmin, colmin);
  chamfer_reduce<<<1, 256, 0, stream>>>(rowmin, colmin, out);
}